// G_GIN_16449724744437
// MI455X (gfx1250) — compile-verified
//
#include <hip/hip_runtime.h>
#include <hip/hip_bf16.h>

#define N_NODES  100000
#define N_EDGES  1600000
#define DIM      128
#define N_GRAPHS 1024
#define BN_EPS   1e-5f

typedef __attribute__((ext_vector_type(2))) float v2f;
typedef __attribute__((ext_vector_type(8))) float v8f;

// ---------------------------------------------------------------------------
// 1) agg = (1 + eps[layer]) * h          (self term of GIN aggregation)
// ---------------------------------------------------------------------------
__global__ void agg_init_kernel(const float* __restrict__ h,
                                float* __restrict__ agg,
                                const float* __restrict__ eps, int layer) {
    size_t i = (size_t)blockIdx.x * blockDim.x + threadIdx.x;   // float4 index
    float s = 1.0f + eps[layer];
    float4 v = ((const float4*)h)[i];
    ((float4*)agg)[i] = make_float4(v.x * s, v.y * s, v.z * s, v.w * s);
}

// ---------------------------------------------------------------------------
// 2) agg[dst[e]] += h[src[e]]  — one wave32 per edge, float4 per lane
// ---------------------------------------------------------------------------
__global__ void scatter_add_kernel(const float* __restrict__ h,
                                   float* __restrict__ agg,
                                   const int* __restrict__ src,
                                   const int* __restrict__ dst) {
    int tid  = blockIdx.x * blockDim.x + threadIdx.x;
    int e    = tid >> 5;            // edge id (wave-uniform)
    int c    = (tid & 31) * 4;      // feature chunk
    int s    = src[e];
    int d    = dst[e];
    float4 v = *(const float4*)(h + (size_t)s * DIM + c);
    float* ap = agg + (size_t)d * DIM + c;
    atomicAdd(ap + 0, v.x);
    atomicAdd(ap + 1, v.y);
    atomicAdd(ap + 2, v.z);
    atomicAdd(ap + 3, v.w);
}

// ---------------------------------------------------------------------------
// zero helpers (graph-capture safe, deterministic)
// ---------------------------------------------------------------------------
__global__ void zero_kernel(float* __restrict__ p, int n) {
    int i = blockIdx.x * blockDim.x + threadIdx.x;
    if (i < n) p[i] = 0.0f;
}

// ---------------------------------------------------------------------------
// 3) out = agg @ W^T + b  via V_WMMA_F32_16X16X4_F32
//    One 16x16 output tile per wave; 8 waves/block cover all 128 columns.
//    A layout (16x4 f32): lane lo=lane&15 -> M, hi=lane>>4 -> K pair {0,1}/{2,3}
//    B layout (4x16 f32): lane lo -> N, hi -> K pair; B[k][n] = W[n][k]
// ---------------------------------------------------------------------------
__global__ void __launch_bounds__(256)
gemm_wmma_kernel(const float* __restrict__ A,      // [N_NODES, DIM]
                 const float* __restrict__ W,      // [DIM, DIM] row-major
                 const float* __restrict__ bias,   // [DIM]
                 float* __restrict__ out) {        // [N_NODES, DIM]
    int lane = threadIdx.x & 31;
    int wave = threadIdx.x >> 5;
    int row0 = blockIdx.x * 16;     // node-row tile
    int j0   = wave * 16;           // output-column tile
    int lo   = lane & 15;
    int hi   = lane >> 4;           // 0 or 1

    const float* arow = A + (size_t)(row0 + lo) * DIM + hi * 2;
    const float* wrow = W + (size_t)(j0 + lo) * DIM + hi * 2;

    v8f c = {0.f, 0.f, 0.f, 0.f, 0.f, 0.f, 0.f, 0.f};
#pragma unroll
    for (int k0 = 0; k0 < DIM; k0 += 4) {
        v2f a = *(const v2f*)(arow + k0);
        v2f b = *(const v2f*)(wrow + k0);
        c = __builtin_amdgcn_wmma_f32_16x16x4_f32(
                /*neg_a=*/false, a, /*neg_b=*/false, b,
                /*c_mod=*/(short)0, c, /*reuse_a=*/false, /*reuse_b=*/false);
    }

    // C/D layout: VGPR v -> row (row0 + hi*8 + v), col (j0 + lo)
    float bj = bias[j0 + lo];
    float* op = out + (size_t)(row0 + hi * 8) * DIM + j0 + lo;
#pragma unroll
    for (int v = 0; v < 8; ++v)
        op[(size_t)v * DIM] = c[v] + bj;
}

// ---------------------------------------------------------------------------
// 4a) column sums / sums-of-squares for BatchNorm (atomics into stats[256])
// ---------------------------------------------------------------------------
#define BN_ROWS 512
__global__ void bn_stats_kernel(const float* __restrict__ h,
                                float* __restrict__ stats) {
    int col = threadIdx.x;                  // 128 threads/block
    int r0  = blockIdx.x * BN_ROWS;
    float s = 0.f, ss = 0.f;
    for (int r = 0; r < BN_ROWS; ++r) {
        int row = r0 + r;
        if (row < N_NODES) {
            float v = h[(size_t)row * DIM + col];
            s += v; ss += v * v;
        }
    }
    atomicAdd(&stats[col], s);
    atomicAdd(&stats[DIM + col], ss);
}

// ---------------------------------------------------------------------------
// 4b) in-place normalize (biased var) + gamma/beta + ReLU
// ---------------------------------------------------------------------------
__global__ void bn_relu_kernel(float* __restrict__ h,
                               const float* __restrict__ stats,
                               const float* __restrict__ gamma,
                               const float* __restrict__ beta) {
    size_t i = (size_t)blockIdx.x * blockDim.x + threadIdx.x;
    int col = (int)(i & (DIM - 1));
    const float invN = 1.0f / (float)N_NODES;
    float mean = stats[col] * invN;
    float var  = stats[DIM + col] * invN - mean * mean;
    float inv  = rsqrtf(var + BN_EPS);
    float v = (h[i] - mean) * inv * gamma[col] + beta[col];
    h[i] = v > 0.f ? v : 0.f;
}

// ---------------------------------------------------------------------------
// 5) global mean pool: atomic sums + counts, then divide
// ---------------------------------------------------------------------------
__global__ void pool_scatter_kernel(const float* __restrict__ h,
                                    const int* __restrict__ batch,
                                    float* __restrict__ sums,
                                    float* __restrict__ counts) {
    int tid  = blockIdx.x * blockDim.x + threadIdx.x;
    int node = tid >> 5;
    int c    = (tid & 31) * 4;
    if (node >= N_NODES) return;
    int g = batch[node];
    float4 v = *(const float4*)(h + (size_t)node * DIM + c);
    float* sp = sums + (size_t)g * DIM + c;
    atomicAdd(sp + 0, v.x);
    atomicAdd(sp + 1, v.y);
    atomicAdd(sp + 2, v.z);
    atomicAdd(sp + 3, v.w);
    if ((tid & 31) == 0) atomicAdd(&counts[g], 1.0f);
}

__global__ void pool_final_kernel(const float* __restrict__ sums,
                                  const float* __restrict__ counts,
                                  float* __restrict__ out) {
    int i = blockIdx.x * blockDim.x + threadIdx.x;   // [0, N_GRAPHS*DIM)
    int g = i >> 7;
    float cnt = counts[g];
    out[i] = sums[i] / (cnt > 1.0f ? cnt : 1.0f);
}

// ---------------------------------------------------------------------------
// host launcher
// ---------------------------------------------------------------------------
extern "C" void kernel_launch(void* const* d_in, const int* in_sizes, int n_in,
                              void* d_out, int out_size, void* d_ws, size_t ws_size,
                              hipStream_t stream) {
    const float* x     = (const float*)d_in[0];
    const int*   ei    = (const int*)d_in[1];
    const int*   batch = (const int*)d_in[2];
    const float* Ws[3] = {(const float*)d_in[3], (const float*)d_in[5], (const float*)d_in[7]};
    const float* bs[3] = {(const float*)d_in[4], (const float*)d_in[6], (const float*)d_in[8]};
    const float* eps   = (const float*)d_in[9];
    const float* gamma = (const float*)d_in[10];
    const float* beta  = (const float*)d_in[11];

    const int* src = ei;
    const int* dst = ei + N_EDGES;

    // workspace layout (floats)
    float* bufA  = (float*)d_ws;                       // [N_NODES*DIM]  aggregation
    float* bufB  = bufA + (size_t)N_NODES * DIM;       // [N_NODES*DIM]  layer output
    float* stats = bufB + (size_t)N_NODES * DIM;       // [256] sums / sumsq
    float* psums = stats + 2 * DIM;                    // [N_GRAPHS*DIM]
    float* pcnts = psums + (size_t)N_GRAPHS * DIM;     // [N_GRAPHS]

    const int elems4   = (N_NODES * DIM) / 4;          // 3,200,000
    const int initBlk  = elems4 / 256;                 // 12,500
    const int scatBlk  = (N_EDGES * 32) / 256;         // 200,000
    const int gemmBlk  = N_NODES / 16;                 // 6,250
    const int statBlk  = (N_NODES + BN_ROWS - 1) / BN_ROWS;  // 196
    const int bnBlk    = (N_NODES * DIM) / 256;        // 50,000
    const int poolBlk  = (N_NODES * 32) / 256;         // 12,500
    const int pzN      = N_GRAPHS * DIM + N_GRAPHS;    // 132,096
    const int pzBlk    = (pzN + 255) / 256;

    const float* h = x;
    for (int layer = 0; layer < 3; ++layer) {
        agg_init_kernel<<<initBlk, 256, 0, stream>>>(h, bufA, eps, layer);
        scatter_add_kernel<<<scatBlk, 256, 0, stream>>>(h, bufA, src, dst);
        zero_kernel<<<1, 256, 0, stream>>>(stats, 2 * DIM);
        gemm_wmma_kernel<<<gemmBlk, 256, 0, stream>>>(bufA, Ws[layer], bs[layer], bufB);
        bn_stats_kernel<<<statBlk, DIM, 0, stream>>>(bufB, stats);
        bn_relu_kernel<<<bnBlk, 256, 0, stream>>>(bufB, stats,
                                                  gamma + layer * DIM,
                                                  beta + layer * DIM);
        h = bufB;
    }

    zero_kernel<<<pzBlk, 256, 0, stream>>>(psums, pzN);   // zeros psums and pcnts
    pool_scatter_kernel<<<poolBlk, 256, 0, stream>>>(bufB, batch, psums, pcnts);
    pool_final_kernel<<<(N_GRAPHS * DIM) / 256, 256, 0, stream>>>(psums, pcnts,
                                                                  (float*)d_out);
}